// CepstrumToImpulseResponse_69234872811594
// MI455X (gfx1250) — compile-verified
//
#include <hip/hip_runtime.h>
#include <stdint.h>

#define MP1    256        // CEP_ORDER+1
#define NIR    512        // IR_LENGTH
#define BATCH  131072
#define BLOCK  128        // 4 wave32 per block
#define STRIDE 260        // dwords per thread-row: 16B aligned, optimal LDS banking
#define ROFF   (BLOCK * STRIDE)            // ring region start (dwords)
#define LDS_FLOATS (2 * BLOCK * STRIDE)    // 66560 floats = 266240 B < 320KB

__global__ __launch_bounds__(BLOCK) void cep2ir_kernel(
    const float* __restrict__ c, float* __restrict__ out)
{
  extern __shared__ float smem[];
  float* myA = smem + threadIdx.x * STRIDE;          // A[k] = k*c_k, k=0..255
  float* myR = smem + ROFF + threadIdx.x * STRIDE;   // h ring, slots 0..255
  const int row = blockIdx.x * BLOCK + threadIdx.x;  // BATCH == grid*BLOCK exactly

  // ---- async copy: c[row][0..255] -> myA[0..255], 16B per lane per op ----
  {
    unsigned lbase = (unsigned)(size_t)myA;  // low 32 bits of generic LDS ptr = LDS addr
    const float* g = c + (size_t)row * MP1;
    #pragma unroll 4
    for (int k = 0; k < MP1; k += 4) {
      unsigned loff = lbase + (unsigned)(k * 4);
      unsigned long long ga = (unsigned long long)(size_t)(g + k);
      asm volatile("global_load_async_to_lds_b128 %0, %1, off"
                   :: "v"(loff), "v"(ga) : "memory");
    }
  }

  // ---- zero the h ring while the async copy is in flight ----
  {
    float4 z = make_float4(0.f, 0.f, 0.f, 0.f);
    #pragma unroll 4
    for (int s = 0; s < 256; s += 4) *(float4*)(myR + s) = z;
  }

#if defined(__has_builtin)
#if __has_builtin(__builtin_amdgcn_s_wait_asynccnt)
  __builtin_amdgcn_s_wait_asynccnt(0);
#else
  asm volatile("s_wait_asynccnt 0" ::: "memory");
#endif
#else
  asm volatile("s_wait_asynccnt 0" ::: "memory");
#endif

  // ---- scale pass: A[k] = k * c_k (vectorized in-place) ----
  float c0 = myA[0];
  #pragma unroll 4
  for (int k = 0; k < MP1; k += 4) {
    float4 v = *(float4*)(myA + k);
    v.x *= (float)k; v.y *= (float)(k + 1); v.z *= (float)(k + 2); v.w *= (float)(k + 3);
    *(float4*)(myA + k) = v;
  }
  // (No barriers anywhere: each thread touches only its own LDS rows.)

  float areg[8];                       // A[1..7] cached in regs for the fixup
  #pragma unroll
  for (int k = 1; k < 8; ++k) areg[k] = myA[k];

  // ---- warm-up: h[0..7] entirely in registers ----
  float hprev[8];                      // rolling: h[n0-8..n0-1]
  hprev[0] = expf(c0);
  myR[0] = hprev[0];
  #pragma unroll
  for (int n = 1; n < 8; ++n) {
    float s = 0.f;
    #pragma unroll
    for (int k = 1; k <= n; ++k) s = fmaf(areg[k], hprev[n - k], s);
    hprev[n] = s / (float)n;
    myR[n] = hprev[n];
  }
  float4* o4 = (float4*)(out + (size_t)row * NIR);
  o4[0] = make_float4(hprev[0], hprev[1], hprev[2], hprev[3]);
  o4[1] = make_float4(hprev[4], hprev[5], hprev[6], hprev[7]);

  // ---- main tiles: outputs n0..n0+7 ----
  for (int n0 = 8; n0 < NIR; n0 += 8) {
    float acc[8], inv[8];
    #pragma unroll
    for (int m = 0; m < 8; ++m) acc[m] = 0.f;
    // hoist the 8 independent divides: scheduler overlaps them with the far FMAs
    #pragma unroll
    for (int m = 0; m < 8; ++m) inv[m] = 1.0f / (float)(n0 + m);

    float P[8], Q[8];
    // initial h window: Q = h[n0-8..n0-1], P = h[n0-16..n0-9] (zeros if j<0)
    {
      const int sq = (n0 - 8) & 255, sp = (n0 - 16) & 255;
      float4 q0 = *(const float4*)(myR + sq), q1 = *(const float4*)(myR + sq + 4);
      float4 p0 = *(const float4*)(myR + sp), p1 = *(const float4*)(myR + sp + 4);
      Q[0]=q0.x; Q[1]=q0.y; Q[2]=q0.z; Q[3]=q0.w; Q[4]=q1.x; Q[5]=q1.y; Q[6]=q1.z; Q[7]=q1.w;
      P[0]=p0.x; P[1]=p0.y; P[2]=p0.z; P[3]=p0.w; P[4]=p1.x; P[5]=p1.y; P[6]=p1.z; P[7]=p1.w;
    }

    // far part: k in [8,255], 31 aligned k-tiles of 8.
    // window covers h[lo..lo+15], lo = n0-k0-8; acc[m] += A[k0+u]*h[n0+m-k0-u]
    #pragma unroll
    for (int kt = 0; kt < 31; ++kt) {
      const int k0 = 8 + 8 * kt;
      float4 a0 = *(const float4*)(myA + k0);
      float4 a1 = *(const float4*)(myA + k0 + 4);
      float A8[8] = {a0.x, a0.y, a0.z, a0.w, a1.x, a1.y, a1.z, a1.w};
      #pragma unroll
      for (int u = 0; u < 8; ++u) {
        #pragma unroll
        for (int m = 0; m < 8; ++m) {
          const int i = m - u + 8;                 // 1..15, constant after unroll
          const float hh = (i < 8) ? P[i] : Q[i - 8];
          acc[m] = fmaf(A8[u], hh, acc[m]);
        }
      }
      // rotate window down (renames under full unroll), load next lower 8 taps
      #pragma unroll
      for (int i = 0; i < 8; ++i) Q[i] = P[i];
      if (kt < 30) {
        const int sp = (n0 - k0 - 16) & 255;
        float4 p0 = *(const float4*)(myR + sp), p1 = *(const float4*)(myR + sp + 4);
        P[0]=p0.x; P[1]=p0.y; P[2]=p0.z; P[3]=p0.w; P[4]=p1.x; P[5]=p1.y; P[6]=p1.z; P[7]=p1.w;
      }
    }

    // near part: k = 1..7. Work on unscaled sums s[m] so the serial chain is
    // FMA-only; the (independent) divides were hoisted to the tile top.
    //   s[m] = acc[m] + sum_{k>m} A[k]*hprev[..] + sum_{k<=m} (A[k]*inv[m-k])*s[m-k]
    //   h[n0+m] = s[m] * inv[m]
    float s[8], hv[8];
    #pragma unroll
    for (int m = 0; m < 8; ++m) {
      float b = acc[m];
      #pragma unroll
      for (int k = m + 1; k < 8; ++k)              // prev-tile taps (final h values)
        b = fmaf(areg[k], hprev[8 + m - k], b);
      #pragma unroll
      for (int k = m; k >= 1; --k)                 // in-tile taps; k=1 last -> short chain
        b = fmaf(areg[k] * inv[m - k], s[m - k], b);
      s[m] = b;
      hv[m] = b * inv[m];
      myR[(n0 + m) & 255] = hv[m];
    }
    #pragma unroll
    for (int m = 0; m < 8; ++m) hprev[m] = hv[m];

    o4[(n0 >> 2)]     = make_float4(hv[0], hv[1], hv[2], hv[3]);
    o4[(n0 >> 2) + 1] = make_float4(hv[4], hv[5], hv[6], hv[7]);
  }
}

extern "C" void kernel_launch(void* const* d_in, const int* in_sizes, int n_in,
                              void* d_out, int out_size, void* d_ws, size_t ws_size,
                              hipStream_t stream) {
  (void)in_sizes; (void)n_in; (void)d_ws; (void)ws_size; (void)out_size;
  const float* c = (const float*)d_in[0];
  float* out = (float*)d_out;

  const size_t shmem = (size_t)LDS_FLOATS * sizeof(float);  // 266240 B
  hipFuncSetAttribute((const void*)cep2ir_kernel,
                      hipFuncAttributeMaxDynamicSharedMemorySize, (int)shmem);

  cep2ir_kernel<<<dim3(BATCH / BLOCK), dim3(BLOCK), shmem, stream>>>(c, out);
}